// VladCore_13572096655817
// MI455X (gfx1250) — compile-verified
//
#include <hip/hip_runtime.h>
#include <stdint.h>

// Problem constants (from reference): B=32, D=512, N=4096, K=64
#define B_  32
#define D_  512
#define N_  4096
#define K_  64

#define QCHUNK 64     // inner-dim (n) chunk staged in LDS per iteration
#define NCH    (N_ / QCHUNK)
#define LDP    68     // padded LDS row stride (floats): 68%64=4 -> conflict-free b64 frag reads

typedef float v2f __attribute__((ext_vector_type(2)));
typedef float v8f __attribute__((ext_vector_type(8)));

// CDNA5 async global->LDS copy (VFLAT GV mode). INST_OFFSET is added to BOTH
// the LDS address (VDST vgpr) and the global address (VADDR vgpr pair), so a
// 32B slice = two b128 beats sharing VGPRs with offset:0 / offset:16.
#define ASYNC_LDS_B128(ldsaddr, gaddr)                                          \
    asm volatile("global_load_async_to_lds_b128 %0, %1, off"                    \
                 :: "v"(ldsaddr), "v"(gaddr) : "memory")
#define ASYNC_LDS_B128_OFF16(ldsaddr, gaddr)                                    \
    asm volatile("global_load_async_to_lds_b128 %0, %1, off offset:16"          \
                 :: "v"(ldsaddr), "v"(gaddr) : "memory")
#define WAIT_ASYNCCNT_0()                                                       \
    asm volatile("s_wait_asynccnt 0x0" ::: "memory")

// One workgroup = 512 threads = 16 wave32 waves.
// Computes a 64(k) x 64(d) block of V for one batch b; each wave owns one
// 16x16 tile of the 4x4 tile grid. Inner n is chunked by 64 with async
// global->LDS double buffering overlapped with the WMMA stream.
__global__ __launch_bounds__(512)
void vlad_wmma_f32_kernel(const float* __restrict__ x,      // (B, D, N)
                          const float* __restrict__ a_bar,  // (B, K, N)
                          const float* __restrict__ c,      // (K, D)
                          float* __restrict__ out)          // (D, K, B)
{
    __shared__ float As[2][K_ * LDP];   // A chunk: [buf][k][q]
    __shared__ float Bs[2][64 * LDP];   // B chunk: [buf][p][q]  (p = d within block)
    __shared__ float a_sum_lds[K_];     // sum_n a_bar[b][k][n]

    const int tid  = threadIdx.x;
    const int b    = blockIdx.x >> 3;          // 8 d-blocks per batch
    const int d0   = (blockIdx.x & 7) * 64;

    const int lane = tid & 31;
    const int wave = tid >> 5;                 // 0..15
    const int mi   = wave >> 2;                // k-tile 0..3
    const int pi   = wave & 3;                 // d-tile 0..3
    const int mrow = lane & 15;
    const int half = lane >> 4;
    const int qoff = half << 1;                // frag: lanes 16-31 hold K=2,3

    // Staging assignment: thread t owns row (t/8), 8 contiguous floats at (t%8)*8.
    const int srow = tid >> 3;                 // 0..63
    const int soff = (tid & 7) << 3;           // 0,8,...,56

    // Global source addresses (chunk 0), advanced by QCHUNK*4 bytes per chunk.
    uint64_t ga = (uint64_t)(uintptr_t)(a_bar + ((size_t)b * K_ + srow) * (size_t)N_ + soff);
    uint64_t gx = (uint64_t)(uintptr_t)(x     + ((size_t)b * D_ + (d0 + srow)) * (size_t)N_ + soff);

    // LDS destination byte addresses for this thread's slice, both buffers.
    // (Generic LDS pointer low 32 bits == LDS byte address per aperture rules.)
    uint32_t ldsA[2], ldsB[2];
    ldsA[0] = (uint32_t)(uintptr_t)&As[0][srow * LDP + soff];
    ldsA[1] = (uint32_t)(uintptr_t)&As[1][srow * LDP + soff];
    ldsB[0] = (uint32_t)(uintptr_t)&Bs[0][srow * LDP + soff];
    ldsB[1] = (uint32_t)(uintptr_t)&Bs[1][srow * LDP + soff];

    if (tid < K_) a_sum_lds[tid] = 0.0f;

    v8f acc = {0.f, 0.f, 0.f, 0.f, 0.f, 0.f, 0.f, 0.f};
    float asum_part = 0.0f;

    const int aRowBase = (mi * 16 + mrow) * LDP + qoff;
    const int bRowBase = (pi * 16 + mrow) * LDP + qoff;

    // Prologue: prefetch chunk 0 into buffer 0 (4 async b128 beats per thread).
    ASYNC_LDS_B128      (ldsA[0], ga);
    ASYNC_LDS_B128_OFF16(ldsA[0], ga);
    ASYNC_LDS_B128      (ldsB[0], gx);
    ASYNC_LDS_B128_OFF16(ldsB[0], gx);

    for (int it = 0; it < NCH; ++it) {
        const int cur = it & 1;
        const int nxt = cur ^ 1;

        WAIT_ASYNCCNT_0();     // my chunk `it` beats have landed in LDS
        __syncthreads();       // everyone's landed; everyone done computing it-1

        // Prefetch chunk it+1 into the buffer just freed by compute(it-1).
        ga += QCHUNK * 4;
        gx += QCHUNK * 4;
        if (it + 1 < NCH) {
            ASYNC_LDS_B128      (ldsA[nxt], ga);
            ASYNC_LDS_B128_OFF16(ldsA[nxt], ga);
            ASYNC_LDS_B128      (ldsB[nxt], gx);
            ASYNC_LDS_B128_OFF16(ldsB[nxt], gx);
        }

        // a_sum partial: read back this thread's 8 staged a_bar values.
        {
            float4 s0 = *(const float4*)&As[cur][srow * LDP + soff];
            float4 s1 = *(const float4*)&As[cur][srow * LDP + soff + 4];
            asum_part += (s0.x + s0.y + s0.z + s0.w)
                       + (s1.x + s1.y + s1.z + s1.w);
        }

        // 16 x V_WMMA_F32_16X16X4_F32 over the 64-deep chunk.
        const float* aBase = &As[cur][aRowBase];
        const float* bBase = &Bs[cur][bRowBase];
        #pragma unroll
        for (int qq = 0; qq < QCHUNK; qq += 4) {
            v2f afrag = *(const v2f*)(aBase + qq);
            v2f bfrag = *(const v2f*)(bBase + qq);
            acc = __builtin_amdgcn_wmma_f32_16x16x4_f32(
                      /*neg_a=*/false, afrag,
                      /*neg_b=*/false, bfrag,
                      /*c_mod=*/(short)0, acc,
                      /*reuse_a=*/false, /*reuse_b=*/false);
        }
    }

    // Finish a_sum: one LDS float atomic per thread (8 partials per k-row).
    atomicAdd(&a_sum_lds[srow], asum_part);
    __syncthreads();

    // Epilogue: V[b,k,d] = acc - a_sum[k]*c[k,d]; out layout (D,K,B).
    // C/D layout: VGPR r, lanes 0-15 -> M=r, lanes 16-31 -> M=r+8; N=lane%16.
    const int d = d0 + pi * 16 + mrow;
    #pragma unroll
    for (int r = 0; r < 8; ++r) {
        int k = mi * 16 + half * 8 + r;
        float v = acc[r] - a_sum_lds[k] * c[(size_t)k * D_ + d];
        out[(size_t)d * (K_ * B_) + (size_t)k * B_ + b] = v;
    }
}

extern "C" void kernel_launch(void* const* d_in, const int* in_sizes, int n_in,
                              void* d_out, int out_size, void* d_ws, size_t ws_size,
                              hipStream_t stream) {
    const float* x     = (const float*)d_in[0];   // (B, D, N)
    const float* a_bar = (const float*)d_in[1];   // (B, K, N)
    const float* c     = (const float*)d_in[2];   // (K, D)
    float*       out   = (float*)d_out;           // (D, K, B)

    dim3 grid(B_ * (D_ / 64));   // 256 workgroups
    dim3 block(512);             // 16 wave32 waves
    vlad_wmma_f32_kernel<<<grid, block, 0, stream>>>(x, a_bar, c, out);
}